// WhoTalksAttention_34918084116495
// MI455X (gfx1250) — compile-verified
//
#include <hip/hip_runtime.h>
#include <hip/hip_bf16.h>

typedef float v2f __attribute__((ext_vector_type(2)));
typedef float v8f __attribute__((ext_vector_type(8)));
typedef float f4  __attribute__((ext_vector_type(4)));

// Order-preserving float->uint encoding so we can use integer atomics for min/max.
__device__ __forceinline__ unsigned enc_f32(float f) {
    unsigned u = __float_as_uint(f);
    return (u & 0x80000000u) ? ~u : (u | 0x80000000u);
}
__device__ __forceinline__ float dec_f32(unsigned e) {
    unsigned u = (e & 0x80000000u) ? (e & 0x7FFFFFFFu) : ~e;
    return __uint_as_float(u);
}

// --- Pass 0: init min/max accumulators (workspace is poisoned, must init) ---
__global__ void init_minmax_kernel(unsigned* __restrict__ mxe, unsigned* __restrict__ mne) {
    int i = threadIdx.x;
    if (i < 128) { mxe[i] = 0u; mne[i] = 0xFFFFFFFFu; }
}

// --- Pass 1: qsum[b,h,e] = sum_l queries[b,l,3,h,e] ------------------------
// One block per (b,h); 256 threads = 4 l-groups x 64 e-lanes.
__global__ void qsum_kernel(const float* __restrict__ q, float* __restrict__ qsum) {
    __shared__ float red[256];
    const int bh = blockIdx.x;            // 0..127
    const int b  = bh >> 3, h = bh & 7;
    const int e  = threadIdx.x & 63;
    const int lq = threadIdx.x >> 6;      // 0..3
    // offset(b,l,3,h,e) = (((b*256 + l)*4 + 3)*8 + h)*64 + e
    const size_t base  = ((((size_t)b * 256) * 4 + 3) * 8 + (size_t)h) * 64 + e;
    const size_t lstep = (size_t)4 * 8 * 64;   // 2048 floats per l
    float s = 0.f;
    for (int l = lq; l < 256; l += 4) s += q[base + (size_t)l * lstep];
    red[threadIdx.x] = s;
    __syncthreads();
    if (lq == 0)
        qsum[bh * 64 + e] = red[e] + red[64 + e] + red[128 + e] + red[192 + e];
}

// --- Pass 2: scores via V_WMMA_F32_16X16X4_F32 + per-(b,h) min/max ---------
// Each wave computes 16 rows: D(16x16) = sum_{t=0..15} A(16x4) x B(4x16),
// with B column 0 = qsum chunk (other columns zero). Column 0 of D lives in
// lanes 0 (rows 0..7) and 16 (rows 8..15), VGPRs 0..7.
// B fragments are built BRANCHLESSLY: every lane loads the qsum pair its
// lane-half would need and multiplies by a 0/1 mask; all 16 pairs are
// hoisted out of the WMMA loop so the steady state is load_b64 + wmma only.
__global__ void scores_wmma_kernel(const float* __restrict__ keys,
                                   const float* __restrict__ qsum,
                                   float* __restrict__ scores,
                                   unsigned* __restrict__ mxe,
                                   unsigned* __restrict__ mne) {
    const int wave = threadIdx.x >> 5;    // 0..3
    const int lane = threadIdx.x & 31;
    const int bh   = blockIdx.x >> 7;     // 128 tile-groups per (b,h)
    const int tg   = blockIdx.x & 127;
    const int b    = bh >> 3, h = bh & 7;
    const int r0   = (tg * 4 + wave) * 16;

    const int m  = lane & 15;
    const int kb = (lane >> 4) * 2;       // lanes 0-15 -> K{0,1}, lanes 16-31 -> K{2,3}
    const float msk = (m == 0) ? 1.0f : 0.0f;   // only column-0 carrier lanes keep qsum

    // Preload + mask all 16 B fragments (branchless, uniform control flow).
    const float* qs = qsum + (size_t)bh * 64 + kb;
    v2f bq[16];
#pragma unroll
    for (int t = 0; t < 16; ++t) {
        v2f p = *(const v2f*)(qs + 4 * t);      // global_load_b64, cached
        bq[t].x = p.x * msk;
        bq[t].y = p.y * msk;
    }

    // keys offset(b,r,h,e) = ((b*8192 + r)*8 + h)*64 + e ; row stride 512 floats
    const float* arow = keys + (((size_t)b * 8192 + (size_t)(r0 + m)) * 8 + (size_t)h) * 64 + kb;

    v8f c = {};
#pragma unroll
    for (int t = 0; t < 16; ++t) {
        v2f a = __builtin_nontemporal_load((const v2f*)(arow + 4 * t));
        c = __builtin_amdgcn_wmma_f32_16x16x4_f32(false, a, false, bq[t],
                                                  (short)0, c, false, false);
    }

    const float scale = 0.125f;           // 1/sqrt(64)
    float lmx = -3.402823466e38f, lmn = 3.402823466e38f;
    if (m == 0) {                          // lanes 0 and 16 hold D column 0
        const int rbase = r0 + ((lane >> 4) * 8);
        float* srow = scores + (size_t)bh * 8192 + rbase;
#pragma unroll
        for (int i = 0; i < 8; ++i) {
            float v = c[i] * scale;
            srow[i] = v;                   // regular store: keep scores hot in L2
            lmx = fmaxf(lmx, v);
            lmn = fminf(lmn, v);
        }
    }
    // combine lanes 0 <-> 16, then one atomic pair per wave
    float omx = __shfl_xor(lmx, 16, 32);
    float omn = __shfl_xor(lmn, 16, 32);
    if (lane == 0) {
        atomicMax(mxe + bh, enc_f32(fmaxf(lmx, omx)));
        atomicMin(mne + bh, enc_f32(fminf(lmn, omn)));
    }
}

// --- Pass 3: out = values * attn, pure streaming (NT b128 loads/stores) ----
__global__ void scale_values_kernel(const float* __restrict__ values,
                                    const float* __restrict__ scores,
                                    const unsigned* __restrict__ mxe,
                                    const unsigned* __restrict__ mne,
                                    float* __restrict__ out) {
    const size_t i4  = (size_t)blockIdx.x * blockDim.x + threadIdx.x; // float4 index
    const size_t seg = i4 >> 4;               // (b*8192 + r)*8 + h  (one seg = 64 floats)
    const int h = (int)(seg & 7);
    const size_t t = seg >> 3;
    const int r = (int)(t & 8191);
    const int b = (int)(t >> 13);
    const int bh = b * 8 + h;

    const float sc = scores[(size_t)bh * 8192 + r];
    const float mx = dec_f32(mxe[bh]);
    const float mn = dec_f32(mne[bh]);
    const float attn = (sc - mn) / (mx - mn);

    f4 v = __builtin_nontemporal_load((const f4*)values + i4);
    v *= attn;
    __builtin_nontemporal_store(v, (f4*)out + i4);
}

extern "C" void kernel_launch(void* const* d_in, const int* in_sizes, int n_in,
                              void* d_out, int out_size, void* d_ws, size_t ws_size,
                              hipStream_t stream) {
    const float* queries = (const float*)d_in[0];
    const float* keys    = (const float*)d_in[1];
    const float* values  = (const float*)d_in[2];
    float* out = (float*)d_out;

    unsigned char* ws = (unsigned char*)d_ws;
    unsigned* mxe = (unsigned*)ws;                 // 128 u32
    unsigned* mne = mxe + 128;                     // 128 u32
    float* qsum   = (float*)(mne + 128);           // 128*64 floats
    float* scores = qsum + 128 * 64;               // 16*8*8192 floats (~4 MB)

    init_minmax_kernel<<<1, 128, 0, stream>>>(mxe, mne);
    qsum_kernel<<<128, 256, 0, stream>>>(queries, qsum);
    // B*H=128 (b,h) pairs * (8192 rows / (4 waves * 16 rows)) = 16384 blocks
    scores_wmma_kernel<<<16384, 128, 0, stream>>>(keys, qsum, scores, mxe, mne);
    // 67,108,864 floats / 4 per thread / 256 threads = 65536 blocks
    scale_values_kernel<<<65536, 256, 0, stream>>>(values, scores, mxe, mne, out);
}